// GraphEmbeddingModel_58583353917443
// MI455X (gfx1250) — compile-verified
//
#include <hip/hip_runtime.h>

// ---------------------------------------------------------------------------
// MI455X (gfx1250, wave32):
//   h1 = relu(x@W1+b1); h1 = topk_mask(h1); h2 = relu(h1@W2+b2);
//   h2 = topk_mask(h2); out = h2@W3+b3
// GEMMs: V_WMMA_F32_16X16X4_F32 (fp32 exact path, fp32 accumulate).
// Global top-50% magnitude mask: fused 4096-bin histogram over float bit
// patterns (activations >= 0 after ReLU => bits monotone in value), top-down
// scan picks the threshold, mask fused into the next layer's LDS staging.
// Software-pipelined: next K-chunk's global loads staged in registers while
// WMMAs consume the current LDS tiles (LOADcnt hides HBM/L2 latency).
// ---------------------------------------------------------------------------

typedef __attribute__((ext_vector_type(2))) float v2f;
typedef __attribute__((ext_vector_type(4))) float v4f;
typedef __attribute__((ext_vector_type(8))) float v8f;

#define HIST_BINS 4096   // top 12 bits of non-negative float => monotone key

// ---------------------------------------------------------------------------
// Fused GEMM:  out[M,N] = act( mask(A[M,K]) @ W[K,N] + bias[N] )
// BM x BN block tile, 256 threads = 8 wave32s, each wave -> 16 x 64 strip.
// WMMA f32 16x16x4: A frag = v2f (lane<16: K=k,k+1 ; lane>=16: K=k+2,k+3),
// B frag = v2f from W^T staged in LDS (same K split per half-wave).
// ---------------------------------------------------------------------------
template <int BM, int BN, int K, bool RELU, bool DO_HIST, bool MASK>
__global__ __launch_bounds__(256)
void gemm_wmma_f32(const float* __restrict__ A,
                   const float* __restrict__ W,
                   const float* __restrict__ bias,
                   float* __restrict__ out,
                   unsigned* __restrict__ hist,
                   const float* __restrict__ thr_in,
                   int M, int N)
{
    constexpr int BK  = 32;
    constexpr int PAD = 4;            // stride 36: lane l bank = (36*l+k)%64 -> conflict-free frags
    constexpr int LDA = BK + PAD;
    constexpr int WAVES_M = BM / 16;
    constexpr int WAVES_N = BN / 64;
    constexpr int A_REGS  = (BM * BK) / (256 * 4);   // v4f per thread for A tile
    constexpr int B_REGS  = (BK * BN) / 256;         // floats per thread for B tile
    static_assert(WAVES_M * WAVES_N == 8, "block must be 8 waves");
    static_assert(K % BK == 0, "K must be a multiple of BK");

    __shared__ float As[BM][LDA];                       // A tile (masked)
    __shared__ float Bs[BN][LDA];                       // W tile, transposed: Bs[n][k]
    __shared__ unsigned hbin[DO_HIST ? HIST_BINS : 1];  // block-private histogram

    const int tid  = threadIdx.x;
    const int lane = tid & 31;
    const int wv   = tid >> 5;
    const int l    = lane & 15;       // position within half-wave
    const int hl   = lane >> 4;       // half-wave select (K split / M split)
    const int wm   = wv % WAVES_M;
    const int wn   = wv / WAVES_M;
    const int row0 = blockIdx.x * BM;
    const int nblk = blockIdx.y * BN;

    const float thr = MASK ? thr_in[0] : 0.0f;

    if (DO_HIST) {
        for (int i = tid; i < HIST_BINS; i += 256) hbin[i] = 0u;
    }

    // ---- register staging buffers for the software pipeline ---------------
    v4f   areg[A_REGS];
    float breg[B_REGS];

    auto load_chunk = [&](int k0) {
        #pragma unroll
        for (int t = 0; t < A_REGS; ++t) {
            const int i  = tid * 4 + t * 1024;
            const int r  = i / BK;
            const int c  = i % BK;
            const int gr = row0 + r;
            v4f v = {};
            if (gr < M)
                v = *(const v4f*)(A + (size_t)gr * K + (k0 + c));
            if (MASK) {
                #pragma unroll
                for (int q = 0; q < 4; ++q)
                    v[q] = (v[q] >= thr) ? v[q] : 0.0f;
            }
            areg[t] = v;
        }
        #pragma unroll
        for (int t = 0; t < B_REGS; ++t) {
            const int i  = tid + t * 256;
            const int kk = i / BN;
            const int n  = i % BN;
            breg[t] = W[(size_t)(k0 + kk) * N + (nblk + n)];
        }
    };

    auto commit_chunk = [&]() {
        #pragma unroll
        for (int t = 0; t < A_REGS; ++t) {
            const int i = tid * 4 + t * 1024;
            *(v4f*)&As[i / BK][i % BK] = areg[t];
        }
        #pragma unroll
        for (int t = 0; t < B_REGS; ++t) {
            const int i = tid + t * 256;
            Bs[i % BN][i / BN] = breg[t];
        }
    };

    v8f acc[4] = {};

    load_chunk(0);
    commit_chunk();

    for (int k0 = 0; k0 < K; k0 += BK) {
        __syncthreads();                       // LDS tiles for chunk k0 ready
        if (k0 + BK < K) load_chunk(k0 + BK);  // global loads fly over WMMAs

        #pragma unroll
        for (int k = 0; k < BK; k += 4) {
            const v2f a = *(const v2f*)&As[wm * 16 + l][k + 2 * hl];
            #pragma unroll
            for (int j = 0; j < 4; ++j) {
                const v2f b = *(const v2f*)&Bs[wn * 64 + j * 16 + l][k + 2 * hl];
                acc[j] = __builtin_amdgcn_wmma_f32_16x16x4_f32(
                    /*neg_a=*/false, a, /*neg_b=*/false, b,
                    /*c_mod=*/(short)0, acc[j],
                    /*reuse_a=*/false, /*reuse_b=*/false);
            }
        }

        __syncthreads();                       // all waves done reading LDS
        if (k0 + BK < K) commit_chunk();       // regs -> LDS for next chunk
    }

    // ---- epilogue: bias, ReLU, store, histogram ---------------------------
    // C/D layout: VGPR r, lanes 0-15 -> (M=r, N=l); lanes 16-31 -> (M=r+8, N=l)
    #pragma unroll
    for (int j = 0; j < 4; ++j) {
        const int col = nblk + wn * 64 + j * 16 + l;
        const float bb = bias[col];
        #pragma unroll
        for (int r = 0; r < 8; ++r) {
            const int row = row0 + wm * 16 + hl * 8 + r;
            float v = acc[j][r] + bb;
            if (RELU) v = (v > 0.0f) ? v : 0.0f;
            if (row < M) {
                if (DO_HIST) {
                    out[(size_t)row * N + col] = v;
                    const unsigned bin = __float_as_uint(v) >> 19; // 12-bit key
                    atomicAdd(&hbin[bin], 1u);
                } else {
                    // final output: never re-read -> keep it out of L2's way
                    __builtin_nontemporal_store(v, &out[(size_t)row * N + col]);
                }
            }
        }
    }

    if (DO_HIST) {
        __syncthreads();
        for (int i = tid; i < HIST_BINS; i += 256)
            if (hbin[i]) atomicAdd(&hist[i], hbin[i]);
    }
}

// ---------------------------------------------------------------------------
__global__ void zero_hist_kernel(unsigned* __restrict__ h, int n)
{
    const int i = blockIdx.x * blockDim.x + threadIdx.x;
    if (i < n) h[i] = 0u;
}

// Top-down scan: largest bucket b with count(values in buckets b..4095) >= keep.
// Threshold = lower edge of bucket b (float reconstructed from bits).
__global__ void select_threshold_kernel(const unsigned* __restrict__ hist,
                                        float* __restrict__ thr,
                                        long long keep)
{
    if (blockIdx.x == 0 && threadIdx.x == 0) {
        long long cum = 0;
        int b = HIST_BINS - 1;
        for (; b > 0; --b) {
            cum += (long long)hist[b];
            if (cum >= keep) break;
        }
        thr[0] = __uint_as_float((unsigned)b << 19);
    }
}

// ---------------------------------------------------------------------------
extern "C" void kernel_launch(void* const* d_in, const int* in_sizes, int n_in,
                              void* d_out, int out_size, void* d_ws, size_t ws_size,
                              hipStream_t stream)
{
    const float* x  = (const float*)d_in[0];
    const float* W1 = (const float*)d_in[1];
    const float* b1 = (const float*)d_in[2];
    const float* W2 = (const float*)d_in[3];
    const float* b2 = (const float*)d_in[4];
    const float* W3 = (const float*)d_in[5];
    const float* b3 = (const float*)d_in[6];

    const int D_IN = 256, H1 = 256, H2 = 128, E = 64;
    const int M = in_sizes[0] / D_IN;   // 500000

    // workspace layout
    char* ws = (char*)d_ws;
    unsigned* hist1 = (unsigned*)ws;                       // 4096 u32
    unsigned* hist2 = hist1 + HIST_BINS;                   // 4096 u32
    float* thr1 = (float*)(ws + 2 * HIST_BINS * sizeof(unsigned));
    float* thr2 = thr1 + 1;
    float* h1 = (float*)(ws + 65536);                      // M x 256 f32
    float* h2 = h1 + (size_t)M * H1;                       // M x 128 f32
    float* out = (float*)d_out;                            // M x 64  f32

    const dim3 blk(256);

    zero_hist_kernel<<<(2 * HIST_BINS + 255) / 256, blk, 0, stream>>>(hist1, 2 * HIST_BINS);

    {   // layer 1: h1 = relu(x @ W1 + b1), histogram
        dim3 g((M + 63) / 64, H1 / 128);
        gemm_wmma_f32<64, 128, 256, true, true, false>
            <<<g, blk, 0, stream>>>(x, W1, b1, h1, hist1, nullptr, M, H1);
    }
    {   // threshold 1: keep = n - int(n*0.5)
        const long long n1 = (long long)M * H1;
        select_threshold_kernel<<<1, 32, 0, stream>>>(hist1, thr1, n1 - n1 / 2);
    }
    {   // layer 2: h2 = relu(mask(h1) @ W2 + b2), histogram
        dim3 g((M + 63) / 64, H2 / 128);
        gemm_wmma_f32<64, 128, 256, true, true, true>
            <<<g, blk, 0, stream>>>(h1, W2, b2, h2, hist2, thr1, M, H2);
    }
    {   // threshold 2
        const long long n2 = (long long)M * H2;
        select_threshold_kernel<<<1, 32, 0, stream>>>(hist2, thr2, n2 - n2 / 2);
    }
    {   // layer 3: out = mask(h2) @ W3 + b3
        dim3 g((M + 127) / 128, E / 64);
        gemm_wmma_f32<128, 64, 128, false, false, true>
            <<<g, blk, 0, stream>>>(h2, W3, b3, out, nullptr, thr2, M, E);
    }
}